// SpectralSpikeSSM_40192303956590
// MI455X (gfx1250) — compile-verified
//
#include <hip/hip_runtime.h>

typedef __attribute__((ext_vector_type(2))) float v2f;
typedef __attribute__((ext_vector_type(8))) float v8f;

#define BATCH 4096
#define WIDTH 2048
#define SD 16

// ---------------------------------------------------------------------------
// Preprocess (single wave): spectral-radius normalize A, compute cb = B·C.
// ws layout: ws[0..255] = A_n (row-major 16x16), ws[256] = cb
// ---------------------------------------------------------------------------
__global__ void SpectralSpikeSSM_prep(const float* __restrict__ A,
                                      const float* __restrict__ B,
                                      const float* __restrict__ C,
                                      float* __restrict__ ws) {
    if (threadIdx.x != 0 || blockIdx.x != 0) return;

    float a[SD][SD];
    for (int i = 0; i < SD; ++i)
        for (int j = 0; j < SD; ++j)
            a[i][j] = A[i * SD + j];

    // Power iteration for dominant |eigenvalue| (spectral radius estimate).
    float x[SD];
    for (int i = 0; i < SD; ++i) x[i] = 1.0f;
    float rad = 0.0f;
    for (int it = 0; it < 64; ++it) {
        float y[SD];
        float nrm2 = 0.0f;
        for (int i = 0; i < SD; ++i) {
            float s = 0.0f;
            for (int j = 0; j < SD; ++j) s = fmaf(a[i][j], x[j], s);
            y[i] = s;
            nrm2 = fmaf(s, s, nrm2);
        }
        float nrm = __builtin_sqrtf(nrm2);
        if (nrm < 1e-30f) { rad = 0.0f; break; }
        rad = nrm;
        float inv = 1.0f / nrm;
        for (int i = 0; i < SD; ++i) x[i] = y[i] * inv;
    }

    const float scale = (rad > 1.0f) ? (1.0f / rad) : 1.0f;
    for (int i = 0; i < SD * SD; ++i) ws[i] = A[i] * scale;

    float cb = 0.0f;
    for (int s = 0; s < SD; ++s) cb = fmaf(B[s], C[s], cb);
    ws[SD * SD] = cb;
}

// ---------------------------------------------------------------------------
// Main kernel: per 16-row tile, wave 0 computes
//   Hrec = H(16x16) @ A_n^T  via 4x V_WMMA_F32_16X16X4_F32  (K split 4x4)
//   d[m] = Hrec[m,:] . C     via shfl_xor reduction
// then all 8 waves stream: out[b,w] = d[b] + cb * x[b,w]  (float4, coalesced)
// grid = (BATCH/16, WIDTH/COLS_PER_BLOCK), block = 256
// ---------------------------------------------------------------------------
#define COLS_PER_BLOCK 512

__global__ __launch_bounds__(256) void SpectralSpikeSSM_main(
    const float* __restrict__ x,
    const float* __restrict__ h,
    const float* __restrict__ An,     // ws[0..255]
    const float* __restrict__ Cvec,   // 16
    const float* __restrict__ cbp,    // ws + 256
    float* __restrict__ out) {

    __shared__ float d_s[SD];

    const int rowBase = blockIdx.x * SD;
    const int lane    = threadIdx.x & 31;

    if (threadIdx.x < 32) {  // wave 0: EXEC all-ones for WMMA
        const int r    = lane & 15;          // row (A operand) / col N (B,D)
        const int hi   = (lane >> 4) & 1;    // lane half
        const int koff = hi * 2;             // K sub-offset within 4-chunk

        v8f acc = {};
        #pragma unroll
        for (int c = 0; c < 4; ++c) {
            // A operand: H tile, 16x4 f32 layout
            //   lanes 0-15: K = 4c+0, 4c+1 ; lanes 16-31: K = 4c+2, 4c+3
            v2f a, b;
            const float* hrow = h + (size_t)(rowBase + r) * SD + 4 * c + koff;
            a.x = hrow[0];
            a.y = hrow[1];
            // B operand: Bmat[k][n] = An[n][k]  (i.e. A_n^T), n = lane%16
            const float* arow = An + (size_t)r * SD + 4 * c + koff;
            b.x = arow[0];
            b.y = arow[1];
            acc = __builtin_amdgcn_wmma_f32_16x16x4_f32(
                false, a, false, b, (short)0, acc, false, false);
        }

        // d[M] = sum_N Hrec[M][N] * C[N]; D layout: acc[rr] = Hrec[rr+8*hi][r]
        const float cv = Cvec[r];
        #pragma unroll
        for (int rr = 0; rr < 8; ++rr) {
            float v = acc[rr] * cv;
            v += __shfl_xor(v, 1, 32);
            v += __shfl_xor(v, 2, 32);
            v += __shfl_xor(v, 4, 32);
            v += __shfl_xor(v, 8, 32);   // stays within 16-lane half
            if (r == 0) d_s[rr + 8 * hi] = v;
        }
    }
    __syncthreads();

    const float cb      = cbp[0];
    const int   colBase = blockIdx.y * COLS_PER_BLOCK;

    // 16 rows x COLS_PER_BLOCK cols as float4: 2048 float4 / 256 threads = 8 ea
    #pragma unroll
    for (int i = 0; i < (SD * COLS_PER_BLOCK / 4) / 256; ++i) {
        const int idx = threadIdx.x + 256 * i;            // 0..2047
        const int row = idx >> 7;                          // / (COLS_PER_BLOCK/4)
        const int c4  = idx & 127;
        const size_t off = (size_t)(rowBase + row) * WIDTH + colBase;
        const float4 xv = *((const float4*)(x + off) + c4);
        const float  dv = d_s[row];
        float4 ov;
        ov.x = fmaf(cb, xv.x, dv);
        ov.y = fmaf(cb, xv.y, dv);
        ov.z = fmaf(cb, xv.z, dv);
        ov.w = fmaf(cb, xv.w, dv);
        *((float4*)(out + off) + c4) = ov;
    }
}

extern "C" void kernel_launch(void* const* d_in, const int* in_sizes, int n_in,
                              void* d_out, int out_size, void* d_ws, size_t ws_size,
                              hipStream_t stream) {
    const float* x = (const float*)d_in[0];   // (4096, 2048)
    const float* h = (const float*)d_in[1];   // (4096, 16)
    const float* A = (const float*)d_in[2];   // (16, 16)
    const float* B = (const float*)d_in[3];   // (16, 1)
    const float* C = (const float*)d_in[4];   // (1, 16)
    float* out = (float*)d_out;
    float* ws  = (float*)d_ws;                // 257 floats used

    SpectralSpikeSSM_prep<<<1, 32, 0, stream>>>(A, B, C, ws);

    dim3 grid(BATCH / SD, WIDTH / COLS_PER_BLOCK);   // (256, 4)
    SpectralSpikeSSM_main<<<grid, 256, 0, stream>>>(
        x, h, ws, C, ws + SD * SD, out);
}